// ChatModel_29618094473399
// MI455X (gfx1250) — compile-verified
//
#include <hip/hip_runtime.h>
#include <hip/hip_bf16.h>
#include <math.h>

typedef float v2f __attribute__((ext_vector_type(2)));
typedef float v8f __attribute__((ext_vector_type(8)));

#define BATCH 32
#define TLEN 512
#define EMBED 128
#define STATE 64
#define HID 160
#define VOCAB 32000
#define TDEC 128
#define G3H (3*HID)   // 480

__device__ __forceinline__ float sigmoidf_(float x) { return 1.0f / (1.0f + expf(-x)); }
__device__ __forceinline__ float softplusf_(float x) { return (x > 20.0f) ? x : log1pf(expf(x)); }

// ---------------------------------------------------------------- embedding
__global__ void embed_kernel(const int* __restrict__ x,
                             const float* __restrict__ emb,
                             float* __restrict__ h) {
    int idx = blockIdx.x * blockDim.x + threadIdx.x;      // over 32*512*128
    int row = idx >> 7;          // b*512+t
    int e   = idx & 127;
    int tok = x[row];
    h[idx] = emb[tok * EMBED + e];
}

// ------------------------------------------------- LN + in_proj + dt/b/c proj
__global__ void mamba_pre(const float* __restrict__ h,
                          const float* __restrict__ ng, const float* __restrict__ nb,
                          const float* __restrict__ in_w, const float* __restrict__ in_b,
                          const float* __restrict__ dt_w, const float* __restrict__ dt_b,
                          const float* __restrict__ b_w,  const float* __restrict__ b_b,
                          const float* __restrict__ c_w,  const float* __restrict__ c_b,
                          const float* __restrict__ a_log,
                          float* __restrict__ xp, float* __restrict__ decay,
                          float* __restrict__ bt, float* __restrict__ ct) {
    __shared__ float r1[256], r2[256];
    __shared__ float xs[EMBED];
    __shared__ float zs[EMBED];
    int row = blockIdx.x;           // b*512+t
    int tid = threadIdx.x;          // 256
    float v = (tid < EMBED) ? h[row * EMBED + tid] : 0.0f;
    r1[tid] = v; r2[tid] = v * v;
    __syncthreads();
    for (int off = 128; off > 0; off >>= 1) {
        if (tid < off) { r1[tid] += r1[tid + off]; r2[tid] += r2[tid + off]; }
        __syncthreads();
    }
    float mu  = r1[0] * (1.0f / EMBED);
    float var = r2[0] * (1.0f / EMBED) - mu * mu;
    float inv = rsqrtf(var + 1e-5f);
    if (tid < EMBED) xs[tid] = (v - mu) * inv * ng[tid] + nb[tid];
    __syncthreads();
    // proj = xn @ in_w.T + in_b : 256 outputs
    {
        const float* w = in_w + tid * EMBED;
        float acc = in_b[tid];
        #pragma unroll 8
        for (int k = 0; k < EMBED; ++k) acc += xs[k] * w[k];
        if (tid < EMBED) xp[row * EMBED + tid] = acc;
        else             zs[tid - EMBED] = acc;
    }
    __syncthreads();
    if (tid < 3 * STATE) {
        int which = tid >> 6, s = tid & 63;
        const float* w = (which == 0 ? dt_w : (which == 1 ? b_w : c_w)) + s * EMBED;
        float bb = (which == 0 ? dt_b : (which == 1 ? b_b : c_b))[s];
        float acc = bb;
        #pragma unroll 8
        for (int k = 0; k < EMBED; ++k) acc += zs[k] * w[k];
        int o = row * STATE + s;
        if (which == 0) {
            float dt = softplusf_(acc) + 1e-4f;
            decay[o] = expf(-expf(a_log[s]) * dt);
        } else if (which == 1) {
            bt[o] = tanhf(acc);
        } else {
            ct[o] = tanhf(acc);
        }
    }
}

// ------------------------------------------------- depthwise conv3 + tanh
__global__ void conv_kernel(const float* __restrict__ xp,
                            const float* __restrict__ cw, const float* __restrict__ cb,
                            float* __restrict__ xc) {
    int idx = blockIdx.x * blockDim.x + threadIdx.x;  // 32*512*128
    int e = idx & 127;
    int t = (idx >> 7) & 511;
    float xm1 = (t > 0)   ? xp[idx - EMBED] : 0.0f;
    float x0  = xp[idx];
    float xp1 = (t < 511) ? xp[idx + EMBED] : 0.0f;
    const float* w = cw + e * 3;
    xc[idx] = tanhf(w[0] * xm1 + w[1] * x0 + w[2] * xp1 + cb[e]);
}

// ------------------------------------------------- SSM recurrence (parallel over b,s)
__global__ void scan_kernel(const float* __restrict__ decay,
                            const float* __restrict__ bt,
                            float* __restrict__ ct_y) {   // in: ct, out: y = ct*state
    int tid = blockIdx.x * blockDim.x + threadIdx.x;      // 2048
    if (tid >= BATCH * STATE) return;
    int b = tid / STATE, s = tid % STATE;
    float state = 0.0f;
    int base = b * TLEN * STATE + s;
    for (int t = 0; t < TLEN; ++t) {
        int o = base + t * STATE;
        state = state * decay[o] + bt[o];
        ct_y[o] = ct_y[o] * state;
    }
}

// ------------------------------------------------- mix + out_proj + residual
__global__ void mamba_mix(float* __restrict__ h,
                          const float* __restrict__ xc, const float* __restrict__ y,
                          const float* __restrict__ dpar,
                          const float* __restrict__ ow, const float* __restrict__ ob) {
    __shared__ float mix[EMBED + STATE];
    __shared__ float res[EMBED];
    int row = blockIdx.x;
    int tid = threadIdx.x;    // 192
    if (tid < EMBED) {
        mix[tid] = xc[row * EMBED + tid] * dpar[tid];
        res[tid] = h[row * EMBED + tid];
    } else {
        mix[tid] = y[row * STATE + (tid - EMBED)];
    }
    __syncthreads();
    if (tid < EMBED) {
        const float* w = ow + tid * (EMBED + STATE);
        float acc = ob[tid];
        #pragma unroll 8
        for (int k = 0; k < EMBED + STATE; ++k) acc += mix[k] * w[k];
        h[row * EMBED + tid] = res[tid] + acc;
    }
}

// ------------------------------------------------- pool + LN + shared + heads + dec_init
__global__ void pool_heads(const float* __restrict__ h, const int* __restrict__ lengths,
                           const float* __restrict__ pg, const float* __restrict__ pb,
                           const float* __restrict__ sw, const float* __restrict__ sb,
                           const float* __restrict__ iw, const float* __restrict__ ib,
                           const float* __restrict__ stw, const float* __restrict__ stb,
                           const float* __restrict__ cw, const float* __restrict__ cb,
                           const float* __restrict__ opw, const float* __restrict__ opb,
                           const float* __restrict__ diw, const float* __restrict__ dib,
                           float* __restrict__ o_int, float* __restrict__ o_sty,
                           float* __restrict__ o_cap, float* __restrict__ o_op,
                           float* __restrict__ o_h, float* __restrict__ h0) {
    __shared__ float r1[128], r2[128];
    __shared__ float hn[EMBED];
    __shared__ float hs[EMBED];
    int b = blockIdx.x, e = threadIdx.x;  // 32 blocks x 128 threads
    int L = lengths[b]; if (L < 1) L = 1;
    float s = 0.0f;
    for (int t = 0; t < L; ++t) s += h[(b * TLEN + t) * EMBED + e];
    float pooled = s / (float)L;
    r1[e] = pooled; r2[e] = pooled * pooled;
    __syncthreads();
    for (int off = 64; off > 0; off >>= 1) {
        if (e < off) { r1[e] += r1[e + off]; r2[e] += r2[e + off]; }
        __syncthreads();
    }
    float mu = r1[0] * (1.0f / EMBED);
    float var = r2[0] * (1.0f / EMBED) - mu * mu;
    hn[e] = (pooled - mu) * rsqrtf(var + 1e-5f) * pg[e] + pb[e];
    __syncthreads();
    {   // shared hidden (relu)
        const float* w = sw + e * EMBED;
        float acc = sb[e];
        #pragma unroll 8
        for (int k = 0; k < EMBED; ++k) acc += hn[k] * w[k];
        float hv = fmaxf(acc, 0.0f);
        hs[e] = hv;
        o_h[b * EMBED + e] = hv;
    }
    __syncthreads();
    if (e < 80) {   // 4 heads: 32 + 8 + 16 + 24
        const float* w; const float* bb; float* dst; int j;
        if (e < 32)      { j = e;      w = iw  + j * EMBED; bb = ib;  dst = o_int + b * 32 + j; }
        else if (e < 40) { j = e - 32; w = stw + j * EMBED; bb = stb; dst = o_sty + b * 8  + j; }
        else if (e < 56) { j = e - 40; w = cw  + j * EMBED; bb = cb;  dst = o_cap + b * 16 + j; }
        else             { j = e - 56; w = opw + j * EMBED; bb = opb; dst = o_op  + b * 24 + j; }
        float acc = bb[j];
        #pragma unroll 8
        for (int k = 0; k < EMBED; ++k) acc += hs[k] * w[k];
        *dst = acc;
    }
    for (int j = e; j < HID; j += 128) {  // dec_init: tanh
        const float* w = diw + j * EMBED;
        float acc = dib[j];
        #pragma unroll 8
        for (int k = 0; k < EMBED; ++k) acc += hs[k] * w[k];
        h0[b * HID + j] = tanhf(acc);
    }
}

// ------------------------------------------------- gx = dec_emb[resp] @ W_ih.T + b
__global__ void gx_kernel(const int* __restrict__ resp,
                          const float* __restrict__ demb,
                          const float* __restrict__ wih, const float* __restrict__ bih,
                          float* __restrict__ gx) {
    __shared__ float xin[EMBED];
    int row = blockIdx.x;    // b*128+td
    int tid = threadIdx.x;   // 256
    if (tid < EMBED) xin[tid] = demb[resp[row] * EMBED + tid];
    __syncthreads();
    for (int j = tid; j < G3H; j += 256) {
        const float* w = wih + j * EMBED;
        float acc = bih[j];
        #pragma unroll 8
        for (int k = 0; k < EMBED; ++k) acc += xin[k] * w[k];
        gx[row * G3H + j] = acc;
    }
}

// ------------------------------------------------- GRU recurrence, one block per batch
__global__ void gru_kernel(const float* __restrict__ gx,
                           const float* __restrict__ h0,
                           const float* __restrict__ whh, const float* __restrict__ bhh,
                           float* __restrict__ dec) {
    __shared__ float hS[HID];
    __shared__ float gh[G3H];
    int b = blockIdx.x, tid = threadIdx.x;   // 512 threads
    if (tid < HID) hS[tid] = h0[b * HID + tid];
    __syncthreads();
    for (int t = 0; t < TDEC; ++t) {
        if (tid < G3H) {
            const float* w = whh + tid * HID;
            float acc = bhh[tid];
            #pragma unroll 8
            for (int k = 0; k < HID; ++k) acc += hS[k] * w[k];
            gh[tid] = acc;
        }
        __syncthreads();
        float hnew = 0.0f;
        if (tid < HID) {
            const float* g = gx + (b * TDEC + t) * G3H;
            float r  = sigmoidf_(g[tid] + gh[tid]);
            float zg = sigmoidf_(g[HID + tid] + gh[HID + tid]);
            float n  = tanhf(g[2 * HID + tid] + r * gh[2 * HID + tid]);
            hnew = (1.0f - zg) * n + zg * hS[tid];
            dec[(b * TDEC + t) * HID + tid] = hnew;
        }
        __syncthreads();
        if (tid < HID) hS[tid] = hnew;
        __syncthreads();
    }
}

// ------------------------------------------------- vocab GEMM: WMMA f32 16x16x4
// C[4096,32000] = A[4096,160] * W[32000,160]^T + bias
__global__ void __launch_bounds__(128) vocab_gemm(const float* __restrict__ A,
                                                  const float* __restrict__ W,
                                                  const float* __restrict__ bias,
                                                  float* __restrict__ out) {
    __shared__ float2 As[64 * 81];   // row m: 80 k-pairs, pad stride 81
    __shared__ float2 Bs[80 * 16];   // k-pair p: 16 columns
    const int tid  = threadIdx.x;
    const int wave = tid >> 5;
    const int lane = tid & 31;
    const int lane_lo = lane & 15;
    const int hv = lane >> 4;        // half (lanes 16-31 hold K+2/M+8 parts)
    const int n0 = blockIdx.x * 16;
    const int m_base = blockIdx.y * 64;

    const float2* A2 = reinterpret_cast<const float2*>(A);
    const float2* W2 = reinterpret_cast<const float2*>(W);
    for (int i = tid; i < 64 * 80; i += 128) {
        int r = i / 80, c = i % 80;
        As[r * 81 + c] = A2[(size_t)(m_base + r) * 80 + c];
    }
    for (int i = tid; i < 16 * 80; i += 128) {
        int n = i / 80, p = i % 80;
        Bs[p * 16 + n] = W2[(size_t)(n0 + n) * 80 + p];
    }
    __syncthreads();

    const int m0 = wave * 16;
    v8f acc = {0.f, 0.f, 0.f, 0.f, 0.f, 0.f, 0.f, 0.f};
    #pragma unroll
    for (int s = 0; s < 40; ++s) {
        int p = 2 * s + hv;                       // k-pair index: k = 4s + 2*hv
        float2 af = As[(m0 + lane_lo) * 81 + p];  // A[m][k], A[m][k+1]
        float2 bf = Bs[p * 16 + lane_lo];         // B[k][n], B[k+1][n]
        v2f a; a.x = af.x; a.y = af.y;
        v2f bb; bb.x = bf.x; bb.y = bf.y;
        acc = __builtin_amdgcn_wmma_f32_16x16x4_f32(false, a, false, bb,
                                                    (short)0, acc, false, false);
    }

    const int n = n0 + lane_lo;
    const float bn = bias[n];
    #pragma unroll
    for (int r = 0; r < 8; ++r) {
        int m = m_base + m0 + r + 8 * hv;
        out[(size_t)m * VOCAB + n] = acc[r] + bn;
    }
}

// =================================================================== launch
extern "C" void kernel_launch(void* const* d_in, const int* in_sizes, int n_in,
                              void* d_out, int out_size, void* d_ws, size_t ws_size,
                              hipStream_t stream) {
    const int*   x        = (const int*)d_in[0];
    const int*   lengths  = (const int*)d_in[1];
    const int*   resp_in  = (const int*)d_in[2];
    const float* emb      = (const float*)d_in[3];
    const float* m_ng     = (const float*)d_in[4];
    const float* m_nb     = (const float*)d_in[5];
    const float* m_in_w   = (const float*)d_in[6];
    const float* m_in_b   = (const float*)d_in[7];
    const float* m_cw     = (const float*)d_in[8];
    const float* m_cb     = (const float*)d_in[9];
    const float* m_dt_w   = (const float*)d_in[10];
    const float* m_dt_b   = (const float*)d_in[11];
    const float* m_b_w    = (const float*)d_in[12];
    const float* m_b_b    = (const float*)d_in[13];
    const float* m_c_w    = (const float*)d_in[14];
    const float* m_c_b    = (const float*)d_in[15];
    const float* m_a_log  = (const float*)d_in[16];
    const float* m_d      = (const float*)d_in[17];
    const float* m_ow     = (const float*)d_in[18];
    const float* m_ob     = (const float*)d_in[19];
    const float* pool_g   = (const float*)d_in[20];
    const float* pool_b   = (const float*)d_in[21];
    const float* sw       = (const float*)d_in[22];
    const float* sb       = (const float*)d_in[23];
    const float* iw       = (const float*)d_in[24];
    const float* ib       = (const float*)d_in[25];
    const float* stw      = (const float*)d_in[26];
    const float* stb      = (const float*)d_in[27];
    const float* cpw      = (const float*)d_in[28];
    const float* cpb      = (const float*)d_in[29];
    const float* opw      = (const float*)d_in[30];
    const float* opb      = (const float*)d_in[31];
    const float* demb     = (const float*)d_in[32];
    const float* diw      = (const float*)d_in[33];
    const float* dib      = (const float*)d_in[34];
    const float* wih      = (const float*)d_in[35];
    const float* whh      = (const float*)d_in[36];
    const float* bih      = (const float*)d_in[37];
    const float* bhh      = (const float*)d_in[38];
    const float* dow      = (const float*)d_in[39];
    const float* dob      = (const float*)d_in[40];

    // workspace layout (floats)
    float* ws        = (float*)d_ws;
    float* buf_h     = ws;                        // 32*512*128 = 2,097,152
    float* buf_xp    = buf_h     + 2097152;       // 2,097,152
    float* buf_conv  = buf_xp    + 2097152;       // 2,097,152
    float* buf_decay = buf_conv  + 2097152;       // 1,048,576
    float* buf_bt    = buf_decay + 1048576;       // 1,048,576
    float* buf_ct    = buf_bt    + 1048576;       // 1,048,576 (becomes y in-place)
    float* buf_gx    = buf_ct    + 1048576;       // 32*128*480 = 1,966,080
    float* buf_dec   = buf_gx    + 1966080;       // 32*128*160 = 655,360
    float* buf_h0    = buf_dec   + 655360;        // 32*160

    // output layout: intent[32,32], style[32,8], cap[32,16], op[32,24],
    //                resp_logits[32,128,32000], h[32,128]
    float* out       = (float*)d_out;
    float* o_int     = out;
    float* o_sty     = out + 1024;
    float* o_cap     = out + 1280;
    float* o_op      = out + 1792;
    float* o_logits  = out + 2560;
    float* o_h       = out + 2560 + (size_t)BATCH * TDEC * VOCAB;

    embed_kernel<<<8192, 256, 0, stream>>>(x, emb, buf_h);

    for (int i = 0; i < 2; ++i) {
        mamba_pre<<<BATCH * TLEN, 256, 0, stream>>>(
            buf_h, m_ng + i * 128, m_nb + i * 128,
            m_in_w + i * 32768, m_in_b + i * 256,
            m_dt_w + i * 8192, m_dt_b + i * 64,
            m_b_w + i * 8192,  m_b_b + i * 64,
            m_c_w + i * 8192,  m_c_b + i * 64,
            m_a_log + i * 64,
            buf_xp, buf_decay, buf_bt, buf_ct);
        conv_kernel<<<8192, 256, 0, stream>>>(buf_xp, m_cw + i * 384, m_cb + i * 128, buf_conv);
        scan_kernel<<<8, 256, 0, stream>>>(buf_decay, buf_bt, buf_ct);
        mamba_mix<<<BATCH * TLEN, 192, 0, stream>>>(buf_h, buf_conv, buf_ct,
                                                    m_d + i * 128,
                                                    m_ow + i * 24576, m_ob + i * 128);
    }

    pool_heads<<<BATCH, 128, 0, stream>>>(buf_h, lengths, pool_g, pool_b, sw, sb,
                                          iw, ib, stw, stb, cpw, cpb, opw, opb,
                                          diw, dib, o_int, o_sty, o_cap, o_op,
                                          o_h, buf_h0);

    gx_kernel<<<BATCH * TDEC, 256, 0, stream>>>(resp_in, demb, wih, bih, buf_gx);
    gru_kernel<<<BATCH, 512, 0, stream>>>(buf_gx, buf_h0, whh, bhh, buf_dec);

    dim3 gv(VOCAB / 16, (BATCH * TDEC) / 64);   // (2000, 64)
    vocab_gemm<<<gv, 128, 0, stream>>>(buf_dec, dow, dob, o_logits);
}